// SnapshotGraphEncoder_7043746365769
// MI455X (gfx1250) — compile-verified
//
#include <hip/hip_runtime.h>
#include <hip/hip_bf16.h>

typedef __attribute__((ext_vector_type(2))) float v2f;
typedef __attribute__((ext_vector_type(8))) float v8f;

// ---------------------------------------------------------------------------
// Elementwise helpers
// ---------------------------------------------------------------------------
__global__ void zero_kernel(float* __restrict__ p, long long n) {
    long long i = (long long)blockIdx.x * blockDim.x + threadIdx.x;
    if (i < n) p[i] = 0.0f;
}

__global__ void concat_kernel(const float* __restrict__ attr,
                              const float* __restrict__ clust,
                              const float* __restrict__ bidir,
                              const float* __restrict__ exist,
                              float* __restrict__ X,
                              int N, int ATTR, int IN) {
    long long idx = (long long)blockIdx.x * blockDim.x + threadIdx.x;
    if (idx >= (long long)N * IN) return;
    int n = (int)(idx / IN);
    int c = (int)(idx % IN);
    float v;
    if (c < ATTR)            v = attr[(long long)n * ATTR + c];
    else if (c == ATTR)      v = clust[n];
    else if (c == ATTR + 1)  v = bidir[n];
    else                     v = exist[n];
    X[idx] = v;
}

// deg[tgt] += 1 per edge
__global__ void deg_kernel(const int* __restrict__ tgt, float* __restrict__ deg, int E) {
    int e = blockIdx.x * blockDim.x + threadIdx.x;
    if (e < E) atomicAdd(&deg[tgt[e]], 1.0f);
}

// agg[tgt] += h[src]; one wave per edge, 4 floats per lane (128 channels)
__global__ void scatter_add_kernel(const float* __restrict__ h,
                                   const int* __restrict__ src,
                                   const int* __restrict__ tgt,
                                   float* __restrict__ agg,
                                   int E, int Hdim) {
    long long gid = (long long)blockIdx.x * blockDim.x + threadIdx.x;
    int e = (int)(gid >> 5);
    int lane = (int)(gid & 31);
    if (e >= E) return;
    int s = src[e];
    int t = tgt[e];
    const float4 v = *(const float4*)(h + (long long)s * Hdim + lane * 4);
    float* base = agg + (long long)t * Hdim + lane * 4;
    atomicAdd(base + 0, v.x);
    atomicAdd(base + 1, v.y);
    atomicAdd(base + 2, v.z);
    atomicAdd(base + 3, v.w);
}

// agg[n][c] /= max(deg[n], 1)   (in place -> neighbor mean)
__global__ void mean_kernel(float* __restrict__ agg, const float* __restrict__ deg,
                            int N, int Hdim) {
    long long idx = (long long)blockIdx.x * blockDim.x + threadIdx.x;
    if (idx >= (long long)N * Hdim) return;
    int n = (int)(idx / Hdim);
    agg[idx] = agg[idx] / fmaxf(deg[n], 1.0f);
}

// ---------------------------------------------------------------------------
// WMMA fp32 GEMM:  Y[n,h] (+)= sum_k X[n,k] * W[h,k] + bias[h]
//   - compile-time K: all 32 V_WMMA_F32_16X16X4_F32 fully unrolled, no
//     remainder loop, minimal SALU
//   - A strip cached in registers (K/2 floats per lane) and reused across all
//     Hdim/16 column tiles -> B is the only per-tile memory traffic
//   - epilogue fuses bias, optional accumulate, relu, and exist-column scale
// Requires: Nrows % 16 == 0, K % 4 == 0, Hdim % 16 == 0 (true: 50000/128/128)
// ---------------------------------------------------------------------------
template <int K>
__global__ void __launch_bounds__(256)
wmma_gemm_kernel(const float* __restrict__ X, const float* __restrict__ W,
                 const float* __restrict__ bias, const float* __restrict__ exist,
                 float* __restrict__ Y, int Hdim, int accum, int relu) {
    constexpr int KS = K / 4;                  // wmma k-steps
    const int lane  = threadIdx.x & 31;
    const int wave  = threadIdx.x >> 5;
    const int waves = blockDim.x >> 5;
    const int tilesN = Hdim >> 4;
    const int row0 = blockIdx.x << 4;          // block owns one 16-row strip

    const int mr = lane & 15;                  // row within tile (A) / col (B,C)
    const int kb = (lane >> 4) << 1;           // k sub-offset: lanes 0-15 -> k0,k1; 16-31 -> k2,k3
    const int mh = (lane >> 4) << 3;           // C/D: upper lanes hold M=8..15

    // Cache this wave's A strip in registers: lane holds A[mr][4i+kb .. +1]
    const float* arow = X + (long long)(row0 + mr) * K;
    v2f areg[KS];
    #pragma unroll
    for (int i = 0; i < KS; ++i) {
        areg[i].x = arow[4 * i + kb];
        areg[i].y = arow[4 * i + kb + 1];
    }

    for (int tn = wave; tn < tilesN; tn += waves) {
        const int col0 = tn << 4;
        const float* brow = W + (long long)(col0 + mr) * K;  // B[k,n] = W[col0+n][k]

        v8f c = {};
        #pragma unroll
        for (int i = 0; i < KS; ++i) {
            v2f b;
            b.x = brow[4 * i + kb];
            b.y = brow[4 * i + kb + 1];
            // D = A(16x4,f32) * B(4x16,f32) + C
            c = __builtin_amdgcn_wmma_f32_16x16x4_f32(
                    false, areg[i], false, b, (short)0, c, false, false);
        }

        const float bcol = bias[col0 + mr];
        #pragma unroll
        for (int v = 0; v < 8; ++v) {
            int row = row0 + mh + v;
            long long off = (long long)row * Hdim + (col0 + mr);
            float val = c[v] + bcol;
            if (accum) val += Y[off];
            if (relu)  val = fmaxf(val, 0.0f);
            if (exist) val *= exist[row];
            Y[off] = val;
        }
    }
}

// ---------------------------------------------------------------------------
// Orchestration
// ---------------------------------------------------------------------------
extern "C" void kernel_launch(void* const* d_in, const int* in_sizes, int n_in,
                              void* d_out, int out_size, void* d_ws, size_t ws_size,
                              hipStream_t stream) {
    const float* attr  = (const float*)d_in[0];
    const float* clust = (const float*)d_in[1];
    const float* bidir = (const float*)d_in[2];
    const float* exist = (const float*)d_in[3];
    const float* W_in  = (const float*)d_in[4];
    const float* b_in  = (const float*)d_in[5];
    const float* W1s   = (const float*)d_in[6];
    const float* b1s   = (const float*)d_in[7];
    const float* W1n   = (const float*)d_in[8];
    const float* b1n   = (const float*)d_in[9];
    const float* W2s   = (const float*)d_in[10];
    const float* b2s   = (const float*)d_in[11];
    const float* W2n   = (const float*)d_in[12];
    const float* b2n   = (const float*)d_in[13];
    const int*   edges = (const int*)d_in[14];

    const int N    = in_sizes[3];            // 50000
    const int ATTR = in_sizes[0] / N;        // 125
    const int IN   = ATTR + 3;               // 128
    const int H    = in_sizes[5];            // 128
    const int OUT  = in_sizes[11];           // 128
    const int E    = in_sizes[14] / 2;       // 800000
    const int* src = edges;
    const int* tgt = edges + E;

    // Workspace layout (floats): X (reused as H1) | H0 | AGG | DEG
    float* X   = (float*)d_ws;
    float* H0  = X  + (long long)N * IN;
    float* AGG = H0 + (long long)N * H;
    float* DEG = AGG + (long long)N * H;
    float* OUTP = (float*)d_out;

    const int TB = 256;
    const long long nNH = (long long)N * H;
    const int gridNH   = (int)((nNH + TB - 1) / TB);
    const int gridE    = (E + TB - 1) / TB;
    const int gridE32  = (int)(((long long)E * 32 + TB - 1) / TB);
    const int gridGemm = N / 16;             // one block per 16-row strip

    // Degree (shared by both layers)
    zero_kernel<<<(N + TB - 1) / TB, TB, 0, stream>>>(DEG, N);
    deg_kernel<<<gridE, TB, 0, stream>>>(tgt, DEG, E);

    // x = concat(...); h0 = relu(x @ W_in^T + b_in)
    concat_kernel<<<gridNH, TB, 0, stream>>>(attr, clust, bidir, exist, X, N, ATTR, IN);
    wmma_gemm_kernel<128><<<gridGemm, TB, 0, stream>>>(X, W_in, b_in, nullptr, H0, H, 0, 1);

    // Layer 1: mean aggregation of h0
    zero_kernel<<<gridNH, TB, 0, stream>>>(AGG, nNH);
    scatter_add_kernel<<<gridE32, TB, 0, stream>>>(H0, src, tgt, AGG, E, H);
    mean_kernel<<<gridNH, TB, 0, stream>>>(AGG, DEG, N, H);

    // h1 = exist * relu(h0 @ W1s^T + b1s + mean @ W1n^T + b1n)   (h1 reuses X)
    wmma_gemm_kernel<128><<<gridGemm, TB, 0, stream>>>(H0,  W1s, b1s, nullptr, X, H, 0, 0);
    wmma_gemm_kernel<128><<<gridGemm, TB, 0, stream>>>(AGG, W1n, b1n, exist,   X, H, 1, 1);

    // Layer 2: mean aggregation of h1
    zero_kernel<<<gridNH, TB, 0, stream>>>(AGG, nNH);
    scatter_add_kernel<<<gridE32, TB, 0, stream>>>(X, src, tgt, AGG, E, H);
    mean_kernel<<<gridNH, TB, 0, stream>>>(AGG, DEG, N, H);

    // out = exist * (h1 @ W2s^T + b2s + mean @ W2n^T + b2n)
    wmma_gemm_kernel<128><<<gridGemm, TB, 0, stream>>>(X,   W2s, b2s, nullptr, OUTP, OUT, 0, 0);
    wmma_gemm_kernel<128><<<gridGemm, TB, 0, stream>>>(AGG, W2n, b2n, exist,   OUTP, OUT, 1, 0);
}